// Scatter_47588237639805
// MI455X (gfx1250) — compile-verified
//
#include <hip/hip_runtime.h>
#include <math.h>

// CDNA5 / gfx1250: wave32, WMMA 16x16x4 f32 for the wavelet-bank combine.
typedef __attribute__((ext_vector_type(2))) float v2f;
typedef __attribute__((ext_vector_type(8))) float v8f;

#define T_STEPS 16
#define INF_CLIP 1e15f

// ---------------------------------------------------------------- CSR build
__global__ void zero_ints_kernel(int* __restrict__ p, int n) {
  int i = blockIdx.x * blockDim.x + threadIdx.x;
  if (i < n) p[i] = 0;
}

__global__ void count_deg_kernel(const int* __restrict__ dst, int* __restrict__ degcnt, int ne) {
  int i = blockIdx.x * blockDim.x + threadIdx.x;
  if (i < ne) atomicAdd(&degcnt[dst[i]], 1);
}

__global__ void block_sum_kernel(const int* __restrict__ degcnt, int* __restrict__ bsum, int n) {
  __shared__ int sh[256];
  int i = blockIdx.x * 256 + threadIdx.x;
  sh[threadIdx.x] = (i < n) ? degcnt[i] : 0;
  __syncthreads();
  for (int off = 128; off > 0; off >>= 1) {
    if (threadIdx.x < off) sh[threadIdx.x] += sh[threadIdx.x + off];
    __syncthreads();
  }
  if (threadIdx.x == 0) bsum[blockIdx.x] = sh[0];
}

__global__ void scan_bsum_kernel(int* __restrict__ bsum, int nb) {
  if (blockIdx.x == 0 && threadIdx.x == 0) {
    int run = 0;
    for (int i = 0; i < nb; ++i) { int v = bsum[i]; bsum[i] = run; run += v; }
  }
}

__global__ void scan_final_kernel(const int* __restrict__ degcnt, const int* __restrict__ bsum,
                                  int* __restrict__ rowptr, float* __restrict__ inv_deg, int n) {
  __shared__ int sh[256];
  int i = blockIdx.x * 256 + threadIdx.x;
  int v = (i < n) ? degcnt[i] : 0;
  sh[threadIdx.x] = v;
  __syncthreads();
  for (int off = 1; off < 256; off <<= 1) {
    int add = (threadIdx.x >= off) ? sh[threadIdx.x - off] : 0;
    __syncthreads();
    sh[threadIdx.x] += add;
    __syncthreads();
  }
  if (i < n) {
    int excl = sh[threadIdx.x] - v + bsum[blockIdx.x];
    rowptr[i] = excl;
    inv_deg[i] = (v > 0) ? 1.0f / (float)v : 0.0f;
    if (i == n - 1) rowptr[n] = excl + v;
  }
}

__global__ void fill_csr_kernel(const int* __restrict__ src, const int* __restrict__ dst,
                                const int* __restrict__ rowptr, int* __restrict__ cursor,
                                int* __restrict__ csr_src, int ne) {
  int i = blockIdx.x * blockDim.x + threadIdx.x;
  if (i < ne) {
    int d = dst[i];
    int p = atomicAdd(&cursor[d], 1);
    csr_src[rowptr[d] + p] = src[i];
  }
}

// ----------------------------------------------------- diffusion (gather-sum)
// One wave32 per node; lane owns VEC consecutive features (F = 32*VEC).
// x_next = 0.5*(x + inv_deg[n] * sum_{src in N(n)} x[src])
template <int VEC>
__global__ void diffuse_kernel(const float* __restrict__ xin, float* __restrict__ xout,
                               const int* __restrict__ rowptr, const int* __restrict__ csr_src,
                               const float* __restrict__ inv_deg, int n_nodes) {
  const int F = 32 * VEC;
  int lane = threadIdx.x & 31;
  int node = (blockIdx.x * blockDim.x + threadIdx.x) >> 5;
  int stride = (gridDim.x * blockDim.x) >> 5;
  for (; node < n_nodes; node += stride) {
    int s = rowptr[node];
    int e = rowptr[node + 1];
    float acc[VEC];
#pragma unroll
    for (int v = 0; v < VEC; ++v) acc[v] = 0.0f;
    for (int j = s; j < e; j += 32) {
      int nIdx = (j + lane < e) ? csr_src[j + lane] : 0;   // coalesced index batch
      if (j + 32 < e) __builtin_prefetch(&csr_src[j + 32 + lane], 0, 1);
      int cnt = e - j; if (cnt > 32) cnt = 32;
      for (int i = 0; i < cnt; ++i) {
        int srcn = __shfl(nIdx, i, 32);                    // broadcast neighbor id
        const float* row = xin + (size_t)srcn * F + lane * VEC;
#pragma unroll
        for (int v = 0; v < VEC; ++v) acc[v] += row[v];
      }
    }
    float w = inv_deg[node];
    const float* xr = xin + (size_t)node * F + lane * VEC;
    float* xo = xout + (size_t)node * F + lane * VEC;
#pragma unroll
    for (int v = 0; v < VEC; ++v) xo[v] = 0.5f * (xr[v] + w * acc[v]);
  }
}

// ------------------------------------------------- wavelet combine via WMMA
// acc[w][i] (+)= sum_{j=0..3} W[w, t0+j] * snap_j[i]  for 16-element tiles i.
// A (16x4 f32): lanes 0-15 hold M=lane, K=0/1 in v0/v1; lanes 16-31: K=2/3.
// B (4x16 f32): lanes 0-15 hold rows K=0/1 (N=lane); lanes 16-31 rows K=2/3.
// C/D (16x16 f32): rows 0-7 in lanes 0-15 (VGPR index = row); rows 0..3 used.
__global__ void wavelet_combine_wmma_kernel(
    const float* __restrict__ b0, const float* __restrict__ b1,
    const float* __restrict__ b2, const float* __restrict__ b3,
    const float* __restrict__ Wm, int t0, int addC,
    float* __restrict__ a0, float* __restrict__ a1,
    float* __restrict__ a2, float* __restrict__ a3, int total) {
  int lane = threadIdx.x & 31;
  bool lo = lane < 16;
  int l16 = lane & 15;
  int wave = (blockIdx.x * blockDim.x + threadIdx.x) >> 5;
  int nwaves = (gridDim.x * blockDim.x) >> 5;
  int ntiles = total >> 4;

  v2f A;
  {
    float av0 = 0.0f, av1 = 0.0f;
    if (l16 < 4) {                       // rows 4..15 of A are zero padding
      int kb = lo ? 0 : 2;
      int ta = t0 + kb, tb = t0 + kb + 1;
      av0 = (ta <= T_STEPS) ? Wm[l16 * (T_STEPS + 1) + ta] : 0.0f;
      av1 = (tb <= T_STEPS) ? Wm[l16 * (T_STEPS + 1) + tb] : 0.0f;
    }
    A.x = av0; A.y = av1;
  }
  const float* pbx = lo ? b0 : b2;
  const float* pby = lo ? b1 : b3;

  for (int t = wave; t < ntiles; t += nwaves) {   // wave-uniform loop, EXEC all-1s at WMMA
    int base = t << 4;
    v2f Bv;
    Bv.x = pbx ? pbx[base + l16] : 0.0f;
    Bv.y = pby ? pby[base + l16] : 0.0f;
    v8f Cv = {};
    if (addC && lo) {
      Cv[0] = a0[base + l16]; Cv[1] = a1[base + l16];
      Cv[2] = a2[base + l16]; Cv[3] = a3[base + l16];
    }
    v8f D = __builtin_amdgcn_wmma_f32_16x16x4_f32(
        false, A, false, Bv, (short)0, Cv, false, false);
    if (lo) {
      a0[base + l16] = D[0]; a1[base + l16] = D[1];
      a2[base + l16] = D[2]; a3[base + l16] = D[3];
    }
  }
}

// -------------------------------------------- s1 = |acc1| in [N][C][4] layout
__global__ void build_s1_kernel(const float* __restrict__ a0, const float* __restrict__ a1,
                                const float* __restrict__ a2, const float* __restrict__ a3,
                                float* __restrict__ s1, int nc) {
  int i = blockIdx.x * blockDim.x + threadIdx.x;
  if (i < nc) {
    float4 v = make_float4(fabsf(a0[i]), fabsf(a1[i]), fabsf(a2[i]), fabsf(a3[i]));
    reinterpret_cast<float4*>(s1)[i] = v;
  }
}

// ------------------------------------------------------------ graph bounds
__global__ void init_bounds_kernel(int* __restrict__ gs, int* __restrict__ ge, int nb, int n) {
  int i = blockIdx.x * blockDim.x + threadIdx.x;
  if (i < nb) { gs[i] = n; ge[i] = 0; }
}

__global__ void graph_bounds_kernel(const int* __restrict__ batch, int* __restrict__ gs,
                                    int* __restrict__ ge, int n) {
  int i = blockIdx.x * blockDim.x + threadIdx.x;
  if (i < n) {
    int b = batch[i];
    atomicMin(&gs[b], i);
    atomicMax(&ge[b], i + 1);
  }
}

// ------------------------------------------------------------------ moments
__device__ __forceinline__ float feat_value(int n, int r, int c,
                                            const float* __restrict__ x,
                                            const float* __restrict__ s1,
                                            const float* __restrict__ a1f,
                                            const float* __restrict__ a2f,
                                            const float* __restrict__ a3f) {
  if (r == 0) return x[(size_t)n * 32 + c];
  if (r <= 4) return s1[((size_t)n * 32 + c) * 4 + (r - 1)];
  // FENG pairs (w_new, k_old): 4->(1,0) 8->(2,0) 9->(2,1) 12->(3,0) 13->(3,1) 14->(3,2)
  const int wtab[6] = {1, 2, 2, 3, 3, 3};
  const int ktab[6] = {0, 0, 1, 0, 1, 2};
  int w = wtab[r - 5], k = ktab[r - 5];
  const float* aw = (w == 1) ? a1f : ((w == 2) ? a2f : a3f);
  return fabsf(aw[((size_t)n * 32 + c) * 4 + k]);
}

__global__ void moments_kernel(const float* __restrict__ x, const float* __restrict__ s1,
                               const float* __restrict__ a1f, const float* __restrict__ a2f,
                               const float* __restrict__ a3f,
                               const int* __restrict__ gs, const int* __restrict__ ge,
                               float* __restrict__ out) {
  int b = blockIdx.x;
  int j = threadIdx.x;            // 0..351 : feature = r*32 + c
  int r = j >> 5, c = j & 31;
  int s = gs[b], e = ge[b];
  size_t ob = (size_t)b * 1408;
  if (e <= s) {
    out[ob + j] = 0.0f; out[ob + 352 + j] = 0.0f;
    out[ob + 704 + j] = 0.0f; out[ob + 1056 + j] = -3.0f;
    return;
  }
  float cnt = (float)(e - s);
  float sum = 0.0f;
  for (int n = s; n < e; ++n) sum += feat_value(n, r, c, x, s1, a1f, a2f, a3f);
  float mean = sum / cnt;
  float s2 = 0.0f, s3 = 0.0f, s4 = 0.0f;
  for (int n = s; n < e; ++n) {
    float d = feat_value(n, r, c, x, s1, a1f, a2f, a3f) - mean;
    float d2 = d * d;
    s2 += d2; s3 += d2 * d; s4 += d2 * d2;
  }
  float var = s2 / cnt, m3 = s3 / cnt, m4 = s4 / cnt;
  float skew = (m3 > INF_CLIP || m3 != m3) ? 0.0f : m3;
  float kurt = m4 - 3.0f;
  kurt = (kurt > INF_CLIP || kurt != kurt) ? -3.0f : kurt;
  out[ob + j] = mean;
  out[ob + 352 + j] = var;
  out[ob + 704 + j] = skew;
  out[ob + 1056 + j] = kurt;
}

__global__ void copy_wavelet_kernel(const float* __restrict__ Wm, float* __restrict__ out, int n) {
  int i = blockIdx.x * blockDim.x + threadIdx.x;
  if (i < n) out[i] = Wm[i];
}

// -------------------------------------------------------------------- host
extern "C" void kernel_launch(void* const* d_in, const int* in_sizes, int n_in,
                              void* d_out, int out_size, void* d_ws, size_t ws_size,
                              hipStream_t stream) {
  (void)n_in; (void)ws_size;
  const float* x    = (const float*)d_in[0];
  const int* eidx   = (const int*)d_in[1];
  const int* batch  = (const int*)d_in[2];
  const float* Wm   = (const float*)d_in[3];
  const int N   = in_sizes[2];
  const int E2  = in_sizes[1] / 2;
  const int wlen = in_sizes[3];                   // 68
  const int Bn  = (out_size - wlen) / 1408;       // 128
  const int* srcp = eidx;
  const int* dstp = eidx + E2;

  char* ws = (char*)d_ws;
  size_t off = 0;
  auto alloc = [&](size_t bytes) -> void* {
    void* p = ws + off;
    off += (bytes + 255) & ~(size_t)255;
    return p;
  };
  int nb = (N + 255) / 256;
  int* degcnt  = (int*)alloc((size_t)N * sizeof(int));
  int* rowptr  = (int*)alloc((size_t)(N + 1) * sizeof(int));
  int* cursor  = (int*)alloc((size_t)N * sizeof(int));
  int* bsum    = (int*)alloc((size_t)nb * sizeof(int));
  int* csr_src = (int*)alloc((size_t)E2 * sizeof(int));
  float* inv_deg = (float*)alloc((size_t)N * sizeof(float));
  int* gs = (int*)alloc((size_t)Bn * sizeof(int));
  int* ge = (int*)alloc((size_t)Bn * sizeof(int));
  size_t f2 = (size_t)N * 128;
  float* snap[4]; for (int i = 0; i < 4; ++i) snap[i] = (float*)alloc(f2 * sizeof(float));
  float* acc[4];  for (int i = 0; i < 4; ++i) acc[i]  = (float*)alloc(f2 * sizeof(float));
  float* s1buf = (float*)alloc(f2 * sizeof(float));

  // --- CSR build (once per call) ---
  zero_ints_kernel<<<(N + 255) / 256, 256, 0, stream>>>(degcnt, N);
  zero_ints_kernel<<<(N + 255) / 256, 256, 0, stream>>>(cursor, N);
  count_deg_kernel<<<(E2 + 255) / 256, 256, 0, stream>>>(dstp, degcnt, E2);
  block_sum_kernel<<<nb, 256, 0, stream>>>(degcnt, bsum, N);
  scan_bsum_kernel<<<1, 32, 0, stream>>>(bsum, nb);
  scan_final_kernel<<<nb, 256, 0, stream>>>(degcnt, bsum, rowptr, inv_deg, N);
  fill_csr_kernel<<<(E2 + 255) / 256, 256, 0, stream>>>(srcp, dstp, rowptr, cursor, csr_src, E2);

  const int CMB_BLOCKS = 960, CMB_TPB = 256;
  int dif_blocks = (N * 32 + 255) / 256;          // one wave32 per node

  auto run_cascade = [&](const float* x0, int vec) {
    int total = N * 32 * vec;
    // init: acc[w] = W[w,0] * x0  (snaps 1..3 null -> zero B rows)
    wavelet_combine_wmma_kernel<<<CMB_BLOCKS, CMB_TPB, 0, stream>>>(
        x0, nullptr, nullptr, nullptr, Wm, 0, 0,
        acc[0], acc[1], acc[2], acc[3], total);
    const float* cur = x0;
    for (int g = 0; g < 4; ++g) {
      for (int j = 0; j < 4; ++j) {
        if (vec == 1)
          diffuse_kernel<1><<<dif_blocks, 256, 0, stream>>>(cur, snap[j], rowptr, csr_src, inv_deg, N);
        else
          diffuse_kernel<4><<<dif_blocks, 256, 0, stream>>>(cur, snap[j], rowptr, csr_src, inv_deg, N);
        cur = snap[j];
      }
      // acc[w] += sum_j W[w, g*4+1+j] * snap_j   via v_wmma_f32_16x16x4_f32
      wavelet_combine_wmma_kernel<<<CMB_BLOCKS, CMB_TPB, 0, stream>>>(
          snap[0], snap[1], snap[2], snap[3], Wm, g * 4 + 1, 1,
          acc[0], acc[1], acc[2], acc[3], total);
    }
  };

  // Cascade 1: F=32 over raw x; then s1[n][c][w] = |acc1[w][n,c]|
  run_cascade(x, 1);
  build_s1_kernel<<<(N * 32 + 255) / 256, 256, 0, stream>>>(
      acc[0], acc[1], acc[2], acc[3], s1buf, N * 32);
  // Cascade 2: F=128 over s1
  run_cascade(s1buf, 4);

  // Per-graph moments over contiguous node ranges
  init_bounds_kernel<<<(Bn + 255) / 256, 256, 0, stream>>>(gs, ge, Bn, N);
  graph_bounds_kernel<<<(N + 255) / 256, 256, 0, stream>>>(batch, gs, ge, N);
  moments_kernel<<<Bn, 352, 0, stream>>>(x, s1buf, acc[1], acc[2], acc[3], gs, ge, (float*)d_out);

  // Tuple output: append wavelet matrix passthrough
  copy_wavelet_kernel<<<1, 96, 0, stream>>>(Wm, (float*)d_out + (out_size - wlen), wlen);
}